// VQVAE_58171037057601
// MI455X (gfx1250) — compile-verified
//
#include <hip/hip_runtime.h>
#include <hip/hip_bf16.h>
#include <math.h>

typedef __attribute__((ext_vector_type(2))) float v2f;
typedef __attribute__((ext_vector_type(8))) float v8f;

#define NUMEMB 512
#define LAT    64

// ---------------------------------------------------------------------------
// conv1: (8,1,65536) -> (8,128,32768), k=4 stride=2 same-pad(1,1), ReLU.
// Cin==1 so only 4 MACs/output: plain VALU kernel.
// ---------------------------------------------------------------------------
__global__ __launch_bounds__(256) void conv1_kernel(
    const float* __restrict__ x, const float* __restrict__ W1,
    const float* __restrict__ b1, float* __restrict__ out)
{
    const int TOUT = 32768, TIN = 65536;
    int idx = blockIdx.x * 256 + threadIdx.x;     // ((b*128)+co)*32768 + t
    int t  = idx & (TOUT - 1);
    int co = (idx >> 15) & 127;
    int b  = idx >> 22;
    const float* xb = x + (size_t)b * TIN;
    float s = b1[co];
#pragma unroll
    for (int k = 0; k < 4; ++k) {
        int p = 2 * t + k - 1;
        float xv = (p >= 0 && p < TIN) ? xb[p] : 0.f;
        s = fmaf(W1[co * 4 + k], xv, s);
    }
    out[(size_t)idx] = fmaxf(s, 0.f);
}

// ---------------------------------------------------------------------------
// Forward strided conv as GEMM with V_WMMA_F32_16X16X4_F32.
//   out[b][co][t] = relu(bias[co] + sum_{ci,k} W[co][ci][k]*in[b][ci][2t+k-1])
// GEMM: A[co][ci*4+k]=W, B[ci*4+k][t]=in[ci][2t+k-1]. Wave = 16(co)x16(t)
// tile, Cin chunks of K=4. Weights for the 16-co tile staged in LDS.
// Block = 8 waves covering 8 consecutive t-tiles; grid (TOUT/128, COUT/16, B).
// ---------------------------------------------------------------------------
template <int CIN, int COUT, int TOUT>
__global__ __launch_bounds__(256) void conv_fwd_wmma(
    const float* __restrict__ in,    // (B, CIN, 2*TOUT)
    const float* __restrict__ W,     // (COUT, CIN, 4)
    const float* __restrict__ bias,  // (COUT)
    float* __restrict__ out)         // (B, COUT, TOUT)
{
    const int TIN = 2 * TOUT;
    __shared__ float wlds[CIN * 4 * 16];          // [kk][co_row]
    const int co_base = blockIdx.y * 16;
    const int b = blockIdx.z;
    for (int i = threadIdx.x; i < CIN * 4 * 16; i += 256) {
        int kk = i >> 4, r = i & 15;
        wlds[i] = W[(size_t)(co_base + r) * (CIN * 4) + kk];
    }
    __syncthreads();

    const int lane = threadIdx.x & 31;
    const int wave = threadIdx.x >> 5;
    const int m    = lane & 15;
    const int half = lane >> 4;
    const int koff = half * 2;                    // A/B K-offset per ISA layout
    const int t_base = (blockIdx.x * 8 + wave) * 16;
    const int t = t_base + m;
    const float* inb = in + (size_t)b * CIN * TIN;

    v8f acc = {};
    for (int ci = 0; ci < CIN; ++ci) {
        v2f a;
        a.x = wlds[(ci * 4 + koff + 0) * 16 + m];
        a.y = wlds[(ci * 4 + koff + 1) * 16 + m];
        const float* row = inb + (size_t)ci * TIN;
        int p0 = 2 * t + koff - 1;                // element K=koff, K=koff+1
        v2f bf;
        bf.x = (p0 >= 0 && p0 < TIN) ? row[p0] : 0.f;
        bf.y = (p0 + 1 < TIN) ? row[p0 + 1] : 0.f;
        acc = __builtin_amdgcn_wmma_f32_16x16x4_f32(
            false, a, false, bf, (short)0, acc, false, false);
    }
    float* outb = out + (size_t)b * COUT * TOUT;
#pragma unroll
    for (int r = 0; r < 8; ++r) {
        int co = co_base + r + half * 8;
        float v = acc[r] + bias[co];
        outb[(size_t)co * TOUT + t_base + m] = fmaxf(v, 0.f);
    }
}

// ---------------------------------------------------------------------------
// ConvTranspose1d (k=4, stride=2, pad=1) as parity-split GEMM + WMMA f32.
//   even s=2n : out += W[ci][co][1]*in[ci][n]   + W[ci][co][3]*in[ci][n-1]
//   odd  s=2n+1: out += W[ci][co][0]*in[ci][n+1] + W[ci][co][2]*in[ci][n]
// K = 2*CIN; weights (torch layout (in,out,k)) re-packed per parity in LDS.
// grid (TIN/128, COUT/16, B*2); blockIdx.z = b*2 + parity.
// ---------------------------------------------------------------------------
template <int CIN, int COUT, int TIN>
__global__ __launch_bounds__(256) void deconv_wmma(
    const float* __restrict__ in,    // (B, CIN, TIN)
    const float* __restrict__ W,     // (CIN, COUT, 4)
    const float* __restrict__ bias,  // (COUT)
    float* __restrict__ out)         // (B, COUT, 2*TIN)
{
    const int TOUT = 2 * TIN;
    const int p = blockIdx.z & 1;
    const int b = blockIdx.z >> 1;
    const int co_base = blockIdx.y * 16;
    const int kA = p ? 0 : 1, kB = p ? 2 : 3;     // tap k for j=0, j=1
    const int tA = p ? 1 : 0, tB = p ? 0 : -1;    // t offset for j=0, j=1

    __shared__ float wlds[CIN * 2 * 16];          // [kk=ci*2+j][co_row]
    for (int i = threadIdx.x; i < CIN * 2 * 16; i += 256) {
        int kk = i >> 4, r = i & 15;
        int ci = kk >> 1, j = kk & 1;
        wlds[i] = W[((size_t)ci * COUT + co_base + r) * 4 + (j ? kB : kA)];
    }
    __syncthreads();

    const int lane = threadIdx.x & 31;
    const int wave = threadIdx.x >> 5;
    const int m    = lane & 15;
    const int half = lane >> 4;
    const int n_base = (blockIdx.x * 8 + wave) * 16;
    const int n = n_base + m;
    const float* inb = in + (size_t)b * CIN * TIN;

    v8f acc = {};
    for (int c = 0; c < CIN / 2; ++c) {           // chunk: kk = 4c..4c+3
        int ci = 2 * c + half;
        v2f a;
        a.x = wlds[(4 * c + half * 2 + 0) * 16 + m];
        a.y = wlds[(4 * c + half * 2 + 1) * 16 + m];
        const float* row = inb + (size_t)ci * TIN;
        int q0 = n + tA, q1 = n + tB;
        v2f bf;
        bf.x = (q0 >= 0 && q0 < TIN) ? row[q0] : 0.f;
        bf.y = (q1 >= 0 && q1 < TIN) ? row[q1] : 0.f;
        acc = __builtin_amdgcn_wmma_f32_16x16x4_f32(
            false, a, false, bf, (short)0, acc, false, false);
    }
    float* outb = out + (size_t)b * COUT * TOUT;
    int s = 2 * (n_base + m) + p;
#pragma unroll
    for (int r = 0; r < 8; ++r) {
        int co = co_base + r + half * 8;
        float v = acc[r] + bias[co];
        outb[(size_t)co * TOUT + s] = fmaxf(v, 0.f);
    }
}

// ---------------------------------------------------------------------------
// ||e||^2 per code
// ---------------------------------------------------------------------------
__global__ void enorm_kernel(const float* __restrict__ emb,
                             float* __restrict__ enorm)
{
    int i = blockIdx.x * blockDim.x + threadIdx.x;
    if (i < NUMEMB) {
        float s = 0.f;
#pragma unroll 8
        for (int d = 0; d < LAT; ++d) {
            float v = emb[i * LAT + d];
            s = fmaf(v, v, s);
        }
        enorm[i] = s;
    }
}

// ---------------------------------------------------------------------------
// VQ: distances via WMMA f32 GEMM S = z_tile(16x64) * emb^T, d = ||e||^2-2S
// (||z||^2 is constant per row -> irrelevant for argmin), running argmin over
// 32 code tiles, butterfly min-reduce across 16-lane N groups, gather q,
// write quantized (B,D,T) and deterministic per-block loss partials.
// grid (T/128, B); block = 256 threads (8 waves x 16 t-rows).
// ---------------------------------------------------------------------------
__global__ __launch_bounds__(256) void vq_kernel(
    const float* __restrict__ z,     // (8, 64, 8192)
    const float* __restrict__ emb,   // (512, 64)
    const float* __restrict__ enorm, // (512)
    float* __restrict__ quant,       // (8, 64, 8192)
    float* __restrict__ partial)     // gridDim.y*gridDim.x partials
{
    const int T = 8192;
    __shared__ float elds[NUMEMB * LAT];          // 128 KB codebook
    __shared__ float enlds[NUMEMB];
    __shared__ float lsum[8];
    for (int i = threadIdx.x; i < NUMEMB * LAT; i += 256) elds[i] = emb[i];
    for (int i = threadIdx.x; i < NUMEMB; i += 256) enlds[i] = enorm[i];
    __syncthreads();

    const int lane = threadIdx.x & 31;
    const int wave = threadIdx.x >> 5;
    const int m    = lane & 15;
    const int half = lane >> 4;
    const int koff = half * 2;
    const int b = blockIdx.y;
    const int t_base = (blockIdx.x * 8 + wave) * 16;
    const float* zb = z + (size_t)b * LAT * T;

    // Preload A-fragments (z tile, 16 K-chunks) into registers: reused 32x.
    v2f afrag[16];
#pragma unroll
    for (int c = 0; c < 16; ++c) {
        afrag[c].x = zb[(size_t)(4 * c + koff + 0) * T + t_base + m];
        afrag[c].y = zb[(size_t)(4 * c + koff + 1) * T + t_base + m];
    }

    float best[8];
    int   bidx[8];
#pragma unroll
    for (int r = 0; r < 8; ++r) { best[r] = 3.4e38f; bidx[r] = 0; }

    for (int jt = 0; jt < NUMEMB / 16; ++jt) {
        int code = jt * 16 + m;                   // this lane's N column
        v8f acc = {};
#pragma unroll
        for (int c = 0; c < 16; ++c) {
            v2f bf;                               // B[d][j] = emb[code][d]
            bf.x = elds[code * LAT + 4 * c + koff + 0];
            bf.y = elds[code * LAT + 4 * c + koff + 1];
            acc = __builtin_amdgcn_wmma_f32_16x16x4_f32(
                false, afrag[c], false, bf, (short)0, acc, false, false);
        }
        float en = enlds[code];
#pragma unroll
        for (int r = 0; r < 8; ++r) {
            float cand = fmaf(-2.0f, acc[r], en);
            if (cand < best[r]) { best[r] = cand; bidx[r] = code; }
        }
    }
    // Min-reduce across the 16 lanes of each N group (ties -> lower index).
#pragma unroll
    for (int off = 8; off >= 1; off >>= 1) {
#pragma unroll
        for (int r = 0; r < 8; ++r) {
            float ov = __shfl_xor(best[r], off, 16);
            int   oi = __shfl_xor(bidx[r], off, 16);
            if (ov < best[r] || (ov == best[r] && oi < bidx[r])) {
                best[r] = ov; bidx[r] = oi;
            }
        }
    }

    // Gather q = emb[idx], write quantized (B,D,T), accumulate loss partial.
    float* qb = quant + (size_t)b * LAT * T;
    float lacc = 0.f;
#pragma unroll
    for (int r = 0; r < 8; ++r) {
        int t = t_base + r + half * 8;            // rows: half 0 -> M 0..7
        int code = bidx[r];
        for (int d = m; d < LAT; d += 16) {
            float qv = elds[code * LAT + d];
            float zv = zb[(size_t)d * T + t];
            float df = qv - zv;
            lacc = fmaf(df, df, lacc);
            qb[(size_t)d * T + t] = qv;
        }
    }
#pragma unroll
    for (int off = 16; off >= 1; off >>= 1) lacc += __shfl_xor(lacc, off, 32);
    if (lane == 0) lsum[wave] = lacc;
    __syncthreads();
    if (threadIdx.x == 0) {
        float s = 0.f;
        for (int w2 = 0; w2 < 8; ++w2) s += lsum[w2];   // fixed order
        partial[blockIdx.y * gridDim.x + blockIdx.x] = s;
    }
}

// Deterministic fixed-order final reduce: 512 partials -> loss scalar.
// loss = (q_loss + COMMIT*e_loss) = 1.25 * sum(diff^2)/ (8*8192*64)
__global__ __launch_bounds__(256) void loss_reduce_kernel(
    const float* __restrict__ partial, float* __restrict__ loss_out)
{
    __shared__ float sm[256];
    sm[threadIdx.x] = partial[threadIdx.x] + partial[threadIdx.x + 256];
    __syncthreads();
    for (int off = 128; off >= 1; off >>= 1) {
        if (threadIdx.x < off) sm[threadIdx.x] += sm[threadIdx.x + off];
        __syncthreads();
    }
    if (threadIdx.x == 0) *loss_out = sm[0] * (1.25f / 4194304.0f);
}

// ---------------------------------------------------------------------------
// Final deconv 128 -> 1 + tanh (tiny: 2 taps x 128 ch per output).
// ---------------------------------------------------------------------------
__global__ __launch_bounds__(256) void deconv_final_kernel(
    const float* __restrict__ in,    // (8,128,32768)
    const float* __restrict__ W,     // (128,1,4)
    const float* __restrict__ bias,  // (1)
    float* __restrict__ out)         // (8,65536)
{
    __shared__ float wk[4][128];
    for (int i = threadIdx.x; i < 128; i += 256) {
        wk[0][i] = W[i * 4 + 0]; wk[1][i] = W[i * 4 + 1];
        wk[2][i] = W[i * 4 + 2]; wk[3][i] = W[i * 4 + 3];
    }
    __syncthreads();
    const int TIN = 32768, TOUT = 65536;
    int idx = blockIdx.x * 256 + threadIdx.x;
    int s = idx & (TOUT - 1);
    int b = idx >> 16;
    const float* inb = in + (size_t)b * 128 * TIN;
    int p = s & 1, n = s >> 1;
    float acc = bias[0];
    if (p == 0) {      // k=1 @ n, k=3 @ n-1
        for (int ci = 0; ci < 128; ++ci) {
            const float* row = inb + (size_t)ci * TIN;
            float v0 = row[n];
            float v1 = (n >= 1) ? row[n - 1] : 0.f;
            acc = fmaf(wk[1][ci], v0, fmaf(wk[3][ci], v1, acc));
        }
    } else {           // k=0 @ n+1, k=2 @ n
        for (int ci = 0; ci < 128; ++ci) {
            const float* row = inb + (size_t)ci * TIN;
            float v0 = (n + 1 < TIN) ? row[n + 1] : 0.f;
            float v1 = row[n];
            acc = fmaf(wk[0][ci], v0, fmaf(wk[2][ci], v1, acc));
        }
    }
    out[idx] = tanhf(acc);
}

// ---------------------------------------------------------------------------
extern "C" void kernel_launch(void* const* d_in, const int* in_sizes, int n_in,
                              void* d_out, int out_size, void* d_ws, size_t ws_size,
                              hipStream_t stream)
{
    (void)in_sizes; (void)n_in; (void)out_size; (void)ws_size;
    const float* x   = (const float*)d_in[0];
    const float* W1  = (const float*)d_in[1];
    const float* b1  = (const float*)d_in[2];
    const float* W2  = (const float*)d_in[3];
    const float* b2  = (const float*)d_in[4];
    const float* W3  = (const float*)d_in[5];
    const float* b3  = (const float*)d_in[6];
    const float* emb = (const float*)d_in[7];
    const float* D1  = (const float*)d_in[8];
    const float* db1 = (const float*)d_in[9];
    const float* D2  = (const float*)d_in[10];
    const float* db2 = (const float*)d_in[11];
    const float* D3  = (const float*)d_in[12];
    const float* db3 = (const float*)d_in[13];

    float* out   = (float*)d_out;
    float* loss  = out + 524288;          // 8*65536
    float* quant = out + 524289;          // (8,64,8192)

    float* ws      = (float*)d_ws;        // ping-pong activation buffers
    float* Abuf    = ws;                  // 8*128*32768 = 33554432 f32
    float* Bbuf    = Abuf + 33554432;     // 8*256*16384 = 33554432 f32
    float* enormb  = Bbuf + 33554432;     // 512
    float* partial = enormb + 512;        // 512
    float* zbuf    = partial + 512;       // 8*64*8192 = 4194304 f32

    // Encoder
    conv1_kernel<<<131072, 256, 0, stream>>>(x, W1, b1, Abuf);
    conv_fwd_wmma<128, 256, 16384><<<dim3(128, 16, 8), 256, 0, stream>>>(
        Abuf, W2, b2, Bbuf);
    conv_fwd_wmma<256, 64, 8192><<<dim3(64, 4, 8), 256, 0, stream>>>(
        Bbuf, W3, b3, zbuf);

    // Vector quantization
    enorm_kernel<<<2, 256, 0, stream>>>(emb, enormb);
    vq_kernel<<<dim3(64, 8), 256, 0, stream>>>(zbuf, emb, enormb, quant, partial);
    loss_reduce_kernel<<<1, 256, 0, stream>>>(partial, loss);

    // Decoder
    deconv_wmma<64, 256, 8192><<<dim3(64, 16, 16), 256, 0, stream>>>(
        quant, D1, db1, Bbuf);
    deconv_wmma<256, 128, 16384><<<dim3(128, 8, 16), 256, 0, stream>>>(
        Bbuf, D2, db2, Abuf);
    deconv_final_kernel<<<2048, 256, 0, stream>>>(Abuf, D3, db3, out);
}